// MultiheadSelfAttention_55688545960710
// MI455X (gfx1250) — compile-verified
//
#include <hip/hip_runtime.h>
#include <hip/hip_bf16.h>

typedef __attribute__((ext_vector_type(2))) float v2f;
typedef __attribute__((ext_vector_type(4))) float v4f;
typedef __attribute__((ext_vector_type(8))) float v8f;

#define D_MODEL 2048
#define NUM_HEADS 16
#define D_K 128
#define SEQ 2048
#define BATCH 4
#define M_ROWS (BATCH * SEQ) /* 8192 */

// ---------------------------------------------------------------------------
// C[M,N] = A[M,K] * W[N,K]^T, M=8192, N=K=2048 (all four projections).
// 128x128 macro-tile / block of 256 threads (8 waves, 4(M) x 2(N) wave grid,
// each wave owns 2x4 = 8 16x16 WMMA tiles). BK=16 LDS tiles, row stride 20
// floats: float4-aligned coop stores + conflict-free v2f fragment reads.
//
// f32 WMMA fragment layout (per ISA 7.12.2):
//   A 16x4 : lane l (l2=l&15, hi=l>>4), vgpr v -> A[M=l2][K = v + 2*hi]
//   B 4x16 : lane l, vgpr v              -> B[N=l2][K = v + 2*hi]
//   C 16x16: lane l, vgpr v              -> C[M = v + 8*hi][N = l2]
// ---------------------------------------------------------------------------
__global__ __launch_bounds__(256) void gemm_awt(float* __restrict__ C,
                                                const float* __restrict__ A,
                                                const float* __restrict__ W) {
  __shared__ float As[128 * 20];
  __shared__ float Bs[128 * 20];
  const int tid = threadIdx.x;
  const int lane = tid & 31, wave = tid >> 5;
  const int l2 = lane & 15, hi = lane >> 4;
  const int wm = wave & 3, wn = wave >> 2;
  const int m0 = blockIdx.y * 128;
  const int n0 = blockIdx.x * 128;

  v8f acc[2][4];
#pragma unroll
  for (int i = 0; i < 2; i++)
#pragma unroll
    for (int j = 0; j < 4; j++) acc[i][j] = {};

  for (int k0 = 0; k0 < D_MODEL; k0 += 16) {
#pragma unroll
    for (int i = 0; i < 2; i++) {  // 512 float4 per operand / 256 threads
      int f = tid + i * 256;
      int row = f >> 2, c4 = (f & 3) * 4;
      *(v4f*)&As[row * 20 + c4] =
          *(const v4f*)&A[(size_t)(m0 + row) * D_MODEL + k0 + c4];
      *(v4f*)&Bs[row * 20 + c4] =
          *(const v4f*)&W[(size_t)(n0 + row) * D_MODEL + k0 + c4];
    }
    __syncthreads();
#pragma unroll
    for (int kk = 0; kk < 16; kk += 4) {
      v2f af[2], bf[4];
#pragma unroll
      for (int i = 0; i < 2; i++)
        af[i] = *(v2f*)&As[(wm * 32 + i * 16 + l2) * 20 + kk + 2 * hi];
#pragma unroll
      for (int j = 0; j < 4; j++)
        bf[j] = *(v2f*)&Bs[(wn * 64 + j * 16 + l2) * 20 + kk + 2 * hi];
#pragma unroll
      for (int i = 0; i < 2; i++)
#pragma unroll
        for (int j = 0; j < 4; j++)
          acc[i][j] = __builtin_amdgcn_wmma_f32_16x16x4_f32(
              false, af[i], false, bf[j], (short)0, acc[i][j], false, false);
    }
    __syncthreads();
  }

#pragma unroll
  for (int i = 0; i < 2; i++)
#pragma unroll
    for (int j = 0; j < 4; j++) {
      int r0 = m0 + wm * 32 + i * 16 + 8 * hi;
      int c = n0 + wn * 64 + j * 16 + l2;
#pragma unroll
      for (int v = 0; v < 8; v++)
        C[(size_t)(r0 + v) * D_MODEL + c] = acc[i][j][v];
    }
}

// ---------------------------------------------------------------------------
// RoPE applied in place to Q and K ([B,S,H*D_K] layout), one thread per
// (b,s,h,freq-pair). inv_freq = 10000^(-2i/128) = exp(-i * ln(1e4)/64).
// ---------------------------------------------------------------------------
__global__ __launch_bounds__(256) void rope_qk(float* __restrict__ Q,
                                               float* __restrict__ K,
                                               const int* __restrict__ pos) {
  long long idx = (long long)blockIdx.x * blockDim.x + threadIdx.x;
  int i = (int)(idx & 63);
  int h = (int)((idx >> 6) & (NUM_HEADS - 1));
  int s = (int)((idx >> 10) & (SEQ - 1));
  int b = (int)(idx >> 21);

  float p = (float)pos[s];
  float freq = expf((float)i * -0.14391156831212787f);  // ln(10000)/64
  float ang = p * freq;
  float sn, cs;
  sincosf(ang, &sn, &cs);

  size_t base = ((size_t)(b * SEQ + s)) * D_MODEL + h * D_K + 2 * i;
  {
    float e = Q[base], o = Q[base + 1];
    Q[base] = e * cs - o * sn;
    Q[base + 1] = e * sn + o * cs;
  }
  {
    float e = K[base], o = K[base + 1];
    K[base] = e * cs - o * sn;
    K[base + 1] = e * sn + o * cs;
  }
}

// ---------------------------------------------------------------------------
// Causal flash attention. Block = (b, h, 128 q-rows), 8 waves; each wave owns
// a 16-row q stripe with Q fragments in registers (pre-scaled by 1/sqrt(128)).
// K/V 16x128 tiles staged in LDS (stride 132 -> conflict-free B-frag reads).
// Online softmax with __shfl_xor(1,2,4,8) row reductions (rows live in the
// 16-lane halves per the C/D layout). P goes C/D -> A layout through a small
// padded LDS staging buffer, then P*V via WMMA.
// ---------------------------------------------------------------------------
__global__ __launch_bounds__(256) void flash_attn(const float* __restrict__ Q,
                                                  const float* __restrict__ K,
                                                  const float* __restrict__ V,
                                                  float* __restrict__ O) {
  __shared__ float Kt[16 * 132];
  __shared__ float Vt[16 * 132];
  __shared__ float Ps[8][16 * 20];
  const int tid = threadIdx.x;
  const int lane = tid & 31, wave = tid >> 5;
  const int l2 = lane & 15, hi = lane >> 4;
  const int b = blockIdx.z, h = blockIdx.y;
  const int q0 = blockIdx.x * 128;
  const int qw0 = q0 + wave * 16;
  const float scale = 0.08838834764831845f;  // 1/sqrt(128)
  const size_t headoff = (size_t)b * SEQ * D_MODEL + (size_t)h * D_K;

  v2f qf[32];
#pragma unroll
  for (int kd = 0; kd < 32; kd++) {
    v2f q = *(const v2f*)&Q[headoff + (size_t)(qw0 + l2) * D_MODEL + kd * 4 + 2 * hi];
    qf[kd].x = q.x * scale;
    qf[kd].y = q.y * scale;
  }

  v8f o[8];
  float m_[8], l_[8];
#pragma unroll
  for (int t = 0; t < 8; t++) o[t] = {};
#pragma unroll
  for (int v = 0; v < 8; v++) { m_[v] = -1e30f; l_[v] = 0.0f; }

  for (int k0 = 0; k0 < q0 + 128; k0 += 16) {
#pragma unroll
    for (int i = 0; i < 2; i++) {  // 512 float4 per matrix / 256 threads
      int f = tid + i * 256;
      int row = f >> 5, c4 = (f & 31) * 4;
      *(v4f*)&Kt[row * 132 + c4] =
          *(const v4f*)&K[headoff + (size_t)(k0 + row) * D_MODEL + c4];
      *(v4f*)&Vt[row * 132 + c4] =
          *(const v4f*)&V[headoff + (size_t)(k0 + row) * D_MODEL + c4];
    }
    __syncthreads();

    if (k0 <= qw0 + 15) {  // wave-uniform: EXEC stays all-ones for WMMA
      v8f sc = {};
#pragma unroll
      for (int kd = 0; kd < 32; kd++) {
        v2f bf = *(v2f*)&Kt[l2 * 132 + kd * 4 + 2 * hi];
        sc = __builtin_amdgcn_wmma_f32_16x16x4_f32(false, qf[kd], false, bf,
                                                   (short)0, sc, false, false);
      }
      // causal mask (finite sentinel: no -inf - -inf NaNs downstream)
#pragma unroll
      for (int v = 0; v < 8; v++)
        if (k0 + l2 > qw0 + v + 8 * hi) sc[v] = -1e30f;

      float alpha[8];
#pragma unroll
      for (int v = 0; v < 8; v++) {
        float rm = sc[v];
        rm = fmaxf(rm, __shfl_xor(rm, 1, 32));
        rm = fmaxf(rm, __shfl_xor(rm, 2, 32));
        rm = fmaxf(rm, __shfl_xor(rm, 4, 32));
        rm = fmaxf(rm, __shfl_xor(rm, 8, 32));
        float mn = fmaxf(m_[v], rm);
        alpha[v] = __expf(m_[v] - mn);
        float p = __expf(sc[v] - mn);
        float rs = p;
        rs += __shfl_xor(rs, 1, 32);
        rs += __shfl_xor(rs, 2, 32);
        rs += __shfl_xor(rs, 4, 32);
        rs += __shfl_xor(rs, 8, 32);
        l_[v] = l_[v] * alpha[v] + rs;
        m_[v] = mn;
        Ps[wave][(v + 8 * hi) * 20 + l2] = p;  // C/D -> row-major staging
      }
#pragma unroll
      for (int t = 0; t < 8; t++)
#pragma unroll
        for (int v = 0; v < 8; v++) o[t][v] *= alpha[v];

#pragma unroll
      for (int kk = 0; kk < 4; kk++) {
        v2f pf = *(v2f*)&Ps[wave][l2 * 20 + kk * 4 + 2 * hi];  // A layout
#pragma unroll
        for (int t = 0; t < 8; t++) {
          v2f vf;  // B layout: N = d column, K = key index
          vf.x = Vt[(kk * 4 + 2 * hi) * 132 + t * 16 + l2];
          vf.y = Vt[(kk * 4 + 2 * hi + 1) * 132 + t * 16 + l2];
          o[t] = __builtin_amdgcn_wmma_f32_16x16x4_f32(false, pf, false, vf,
                                                       (short)0, o[t], false, false);
        }
      }
    }
    __syncthreads();
  }

  float inv[8];
#pragma unroll
  for (int v = 0; v < 8; v++) inv[v] = 1.0f / l_[v];
#pragma unroll
  for (int t = 0; t < 8; t++)
#pragma unroll
    for (int v = 0; v < 8; v++)
      O[(size_t)(b * SEQ + qw0 + v + 8 * hi) * D_MODEL + h * D_K + t * 16 + l2] =
          o[t][v] * inv[v];
}

extern "C" void kernel_launch(void* const* d_in, const int* in_sizes, int n_in,
                              void* d_out, int out_size, void* d_ws, size_t ws_size,
                              hipStream_t stream) {
  const float* x  = (const float*)d_in[0];
  const float* wq = (const float*)d_in[1];
  const float* wk = (const float*)d_in[2];
  const float* wv = (const float*)d_in[3];
  const float* wo = (const float*)d_in[4];
  const int* pos  = (const int*)d_in[5];
  float* out = (float*)d_out;

  const size_t NQ = (size_t)BATCH * SEQ * D_MODEL;  // 16.7M floats each
  float* Qb = (float*)d_ws;
  float* Kb = Qb + NQ;
  float* Vb = Kb + NQ;
  float* AO = Vb + NQ;

  dim3 gg(D_MODEL / 128, M_ROWS / 128);  // (16, 64)
  gemm_awt<<<gg, 256, 0, stream>>>(Qb, x, wq);
  gemm_awt<<<gg, 256, 0, stream>>>(Kb, x, wk);
  gemm_awt<<<gg, 256, 0, stream>>>(Vb, x, wv);

  long long total = (long long)BATCH * SEQ * NUM_HEADS * 64;  // 8.4M pairs
  rope_qk<<<(unsigned)((total + 255) / 256), 256, 0, stream>>>(Qb, Kb, pos);

  dim3 ga(SEQ / 128, NUM_HEADS, BATCH);  // (16, 16, 4)
  flash_attn<<<ga, 256, 0, stream>>>(Qb, Kb, Vb, AO);

  gemm_awt<<<gg, 256, 0, stream>>>(out, AO, wo);
}